// Interaction_40570261078236
// MI455X (gfx1250) — compile-verified
//
#include <hip/hip_runtime.h>

typedef float v2f __attribute__((ext_vector_type(2)));
typedef float v8f __attribute__((ext_vector_type(8)));

#define CCH 32
#define RBF 320
#define CUTOFF_F 5.0f
#define PI_F 3.14159265358979323846f

#define PHI_S 321   // padded LDS row strides (odd -> conflict-free fragment reads)
#define H_S   65
#define C_S   97

__device__ __forceinline__ float silu_f(float x) {
  return x / (1.0f + __expf(-x));
}

__device__ __forceinline__ void atomic_add_f(float* p, float v) {
  __hip_atomic_fetch_add(p, v, __ATOMIC_RELAXED, __HIP_MEMORY_SCOPE_AGENT);
}

// 3x3 matmul: C = A*B (row-major 9-float)
__device__ __forceinline__ void mm3(const float* A, const float* B, float* C) {
#pragma unroll
  for (int i = 0; i < 3; ++i)
#pragma unroll
    for (int j = 0; j < 3; ++j)
      C[i * 3 + j] = A[i * 3 + 0] * B[0 * 3 + j] +
                     A[i * 3 + 1] * B[1 * 3 + j] +
                     A[i * 3 + 2] * B[2 * 3 + j];
}

// compact comps {t, a01,a02,a12, s00,s01,s02,s11,s12} -> full 3x3
__device__ __forceinline__ void recon3(const float* c, float* m) {
  float t = c[0], a01 = c[1], a02 = c[2], a12 = c[3];
  float s00 = c[4], s01 = c[5], s02 = c[6], s11 = c[7], s12 = c[8];
  m[0] = t + s00;   m[1] = a01 + s01; m[2] = a02 + s02;
  m[3] = s01 - a01; m[4] = t + s11;   m[5] = a12 + s12;
  m[6] = s02 - a02; m[7] = s12 - a12; m[8] = t - s00 - s11;
}

// full 3x3 -> compact comps
__device__ __forceinline__ void decomp3(const float* x, float* c) {
  float t = (x[0] + x[4] + x[8]) * (1.0f / 3.0f);
  c[0] = t;
  c[1] = 0.5f * (x[1] - x[3]);
  c[2] = 0.5f * (x[2] - x[6]);
  c[3] = 0.5f * (x[5] - x[7]);
  c[4] = x[0] - t;
  c[5] = 0.5f * (x[1] + x[3]);
  c[6] = 0.5f * (x[2] + x[6]);
  c[7] = x[4] - t;
  c[8] = 0.5f * (x[5] + x[7]);
}

// ---------------------------------------------------------------------------
// K0: per-atom decompose + pre channel-mix. 4 atoms / 128-thread block.
// Writes compact Part[n][c][9] and pre-mixed Y0c[n][o][9].
// ---------------------------------------------------------------------------
__global__ void __launch_bounds__(128) pre_kernel(
    const float* __restrict__ X,
    const float* __restrict__ WI, const float* __restrict__ WA,
    const float* __restrict__ WS,
    float* __restrict__ Part, float* __restrict__ Y0c, int N) {
  __shared__ float s_part[4 * CCH * 9];
  __shared__ float s_WI[CCH * CCH], s_WA[CCH * CCH], s_WS[CCH * CCH];
  int tid = threadIdx.x;
  // weights transposed into LDS: s_W[c*32 + o] = W[o][c]  (conflict-free reads)
  for (int p = tid; p < CCH * CCH; p += 128) {
    int o = p >> 5, c = p & 31;
    s_WI[c * CCH + o] = WI[p];
    s_WA[c * CCH + o] = WA[p];
    s_WS[c * CCH + o] = WS[p];
  }
  int nl = tid >> 5, ch = tid & 31;
  int n = blockIdx.x * 4 + nl;
  float cp[9];
  if (n < N) {
    const float* x = X + ((size_t)n * CCH + ch) * 9;
    decomp3(x, cp);
    float* gp = Part + ((size_t)n * CCH + ch) * 9;
#pragma unroll
    for (int q = 0; q < 9; ++q) gp[q] = cp[q];
  } else {
#pragma unroll
    for (int q = 0; q < 9; ++q) cp[q] = 0.0f;
  }
  float* sp = &s_part[(nl * CCH + ch) * 9];
#pragma unroll
  for (int q = 0; q < 9; ++q) sp[q] = cp[q];
  __syncthreads();

  // phase B: thread (nl, o): out[o][comp] = sum_c W_sel[o][c] * part[c][comp]
  int o = ch;
  float acc[9] = {0, 0, 0, 0, 0, 0, 0, 0, 0};
  for (int c = 0; c < CCH; ++c) {
    float wi = s_WI[c * CCH + o];
    float wa = s_WA[c * CCH + o];
    float wsv = s_WS[c * CCH + o];
    const float* pc = &s_part[(nl * CCH + c) * 9];  // LDS broadcast across wave
    acc[0] += wi * pc[0];
    acc[1] += wa * pc[1]; acc[2] += wa * pc[2]; acc[3] += wa * pc[3];
    acc[4] += wsv * pc[4]; acc[5] += wsv * pc[5]; acc[6] += wsv * pc[6];
    acc[7] += wsv * pc[7]; acc[8] += wsv * pc[8];
  }
  if (n < N) {
    float* yp = Y0c + ((size_t)n * CCH + o) * 9;
#pragma unroll
    for (int q = 0; q < 9; ++q) yp[q] = acc[q];
  }
}

// ---------------------------------------------------------------------------
// K1: per-edge RBF -> MLP (fp32 WMMA 16x16x4, dual accumulator chains) ->
// envelope -> gather/scatter. One 128-thread block (4 waves) per 16 edges.
// ---------------------------------------------------------------------------
__global__ void __launch_bounds__(128) edge_kernel(
    const float* __restrict__ r, const int* __restrict__ eidx,
    const float* __restrict__ W1, const float* __restrict__ b1,
    const float* __restrict__ W2, const float* __restrict__ b2,
    const float* __restrict__ Part, float* __restrict__ Mb, int E) {
  __shared__ float s_phi[16 * PHI_S];
  __shared__ float s_h[16 * H_S];
  __shared__ float s_c[16 * C_S];
  __shared__ float s_er[16], s_env[16];
  __shared__ int s_i[16], s_j[16];

  int tid = threadIdx.x;
  int e0 = blockIdx.x * 16;

  if (tid < 16) {
    int e = e0 + tid;
    if (e >= E) e = E - 1;
    float rr = r[e];
    s_er[tid] = __expf(-rr);
    s_env[tid] =
        (rr < CUTOFF_F) ? 0.5f * (__cosf(PI_F * rr * (1.0f / CUTOFF_F)) + 1.0f)
                        : 0.0f;
    s_i[tid] = eidx[e];
    s_j[tid] = eidx[E + e];
  }
  __syncthreads();

  // ---- phi tile (16 x 320) into LDS: 8 threads per edge, 40 exps each ----
  {
    const float C0 = 0.006737946999085467f;              // exp(-5)
    const float DC = (1.0f - C0) / (float)(RBF - 1);
    const float SW = (2.0f / (float)RBF) * (1.0f - C0);
    const float BETA = 1.0f / (SW * SW);
    int e = tid >> 3;
    float er = s_er[e];
    float* row = &s_phi[e * PHI_S];
    for (int k = (tid & 7); k < RBF; k += 8) {
      float d = er - (C0 + (float)k * DC);
      row[k] = __expf(-BETA * d * d);
    }
  }
  __syncthreads();

  int wave = tid >> 5;
  int lane = tid & 31;
  int half = lane >> 4;       // 0: K,K+1   1: K+2,K+3
  int ln = lane & 15;         // fragment row (A) / col (B)
  int koff = half * 2;

  // ---- GEMM1: h(16x64) = phi(16x320) @ W1^T, wave w owns cols [16w,16w+16) --
  {
    int colW = wave * 16 + ln;                 // output feature (W1 row)
    const float* w1row = W1 + colW * RBF;
    const float* arow = &s_phi[ln * PHI_S];
    v8f acc0 = {0.f, 0.f, 0.f, 0.f, 0.f, 0.f, 0.f, 0.f};
    v8f acc1 = {0.f, 0.f, 0.f, 0.f, 0.f, 0.f, 0.f, 0.f};
#pragma unroll 4
    for (int k = 0; k < RBF; k += 8) {         // two independent WMMA chains
      v2f a, b;
      a.x = arow[k + koff];
      a.y = arow[k + koff + 1];
      b.x = w1row[k + koff];
      b.y = w1row[k + koff + 1];
      acc0 = __builtin_amdgcn_wmma_f32_16x16x4_f32(false, a, false, b, (short)0,
                                                   acc0, false, false);
      a.x = arow[k + 4 + koff];
      a.y = arow[k + 4 + koff + 1];
      b.x = w1row[k + 4 + koff];
      b.y = w1row[k + 4 + koff + 1];
      acc1 = __builtin_amdgcn_wmma_f32_16x16x4_f32(false, a, false, b, (short)0,
                                                   acc1, false, false);
    }
    float bb = b1[colW];
#pragma unroll
    for (int v = 0; v < 8; ++v) {
      int e = v + half * 8;                    // C/D layout: M = v + 8*half
      s_h[e * H_S + colW] = silu_f(acc0[v] + acc1[v] + bb);
    }
  }
  __syncthreads();

  // ---- GEMM2: c(16x96) = h(16x64) @ W2^T, 6 col-tiles over 4 waves --------
  for (int t = wave; t < 6; t += 4) {
    int col = t * 16 + ln;
    const float* w2row = W2 + col * (2 * CCH);
    const float* arow = &s_h[ln * H_S];
    v8f acc0 = {0.f, 0.f, 0.f, 0.f, 0.f, 0.f, 0.f, 0.f};
    v8f acc1 = {0.f, 0.f, 0.f, 0.f, 0.f, 0.f, 0.f, 0.f};
#pragma unroll
    for (int k = 0; k < 2 * CCH; k += 8) {
      v2f a, b;
      a.x = arow[k + koff];
      a.y = arow[k + koff + 1];
      b.x = w2row[k + koff];
      b.y = w2row[k + koff + 1];
      acc0 = __builtin_amdgcn_wmma_f32_16x16x4_f32(false, a, false, b, (short)0,
                                                   acc0, false, false);
      a.x = arow[k + 4 + koff];
      a.y = arow[k + 4 + koff + 1];
      b.x = w2row[k + 4 + koff];
      b.y = w2row[k + 4 + koff + 1];
      acc1 = __builtin_amdgcn_wmma_f32_16x16x4_f32(false, a, false, b, (short)0,
                                                   acc1, false, false);
    }
    float bb = b2[col];
#pragma unroll
    for (int v = 0; v < 8; ++v) {
      int e = v + half * 8;
      s_c[e * C_S + col] = silu_f(acc0[v] + acc1[v] + bb) * s_env[e];
    }
  }
  __syncthreads();

  // ---- gather Part[j], gate, scatter-add into M[i] (L2-resident atomics) --
  for (int p = tid; p < 16 * CCH; p += 128) {
    int e = p >> 5;
    int ch = p & 31;
    if (e0 + e < E) {
      float fI = s_c[e * C_S + ch];
      float fA = s_c[e * C_S + CCH + ch];
      float fS = s_c[e * C_S + 2 * CCH + ch];
      const float* pj = Part + ((size_t)s_j[e] * CCH + ch) * 9;
      float* mi = Mb + ((size_t)s_i[e] * CCH + ch) * 9;
      atomic_add_f(mi + 0, fI * pj[0]);
      atomic_add_f(mi + 1, fA * pj[1]);
      atomic_add_f(mi + 2, fA * pj[2]);
      atomic_add_f(mi + 3, fA * pj[3]);
      atomic_add_f(mi + 4, fS * pj[4]);
      atomic_add_f(mi + 5, fS * pj[5]);
      atomic_add_f(mi + 6, fS * pj[6]);
      atomic_add_f(mi + 7, fS * pj[7]);
      atomic_add_f(mi + 8, fS * pj[8]);
    }
  }
}

// ---------------------------------------------------------------------------
// K2: per-atom post: Z = Y@M + M@Y, normalize, decompose, post-mix, Y + Y@Y.
// ---------------------------------------------------------------------------
__global__ void __launch_bounds__(128) post_kernel(
    const float* __restrict__ Y0c, const float* __restrict__ Mb,
    const float* __restrict__ WI, const float* __restrict__ WA,
    const float* __restrict__ WS, float* __restrict__ out, int N) {
  __shared__ float s_z[4 * CCH * 9];
  __shared__ float s_WI[CCH * CCH], s_WA[CCH * CCH], s_WS[CCH * CCH];
  int tid = threadIdx.x;
  for (int p = tid; p < CCH * CCH; p += 128) {
    int o = p >> 5, c = p & 31;
    s_WI[c * CCH + o] = WI[p];
    s_WA[c * CCH + o] = WA[p];
    s_WS[c * CCH + o] = WS[p];
  }
  int nl = tid >> 5, ch = tid & 31;
  int n = blockIdx.x * 4 + nl;
  float zc[9];
  if (n < N) {
    float Y[9], M[9], ZA[9], ZB[9], Z[9];
    recon3(Y0c + ((size_t)n * CCH + ch) * 9, Y);
    recon3(Mb + ((size_t)n * CCH + ch) * 9, M);
    mm3(Y, M, ZA);
    mm3(M, Y, ZB);
    float fr = 0.0f;
#pragma unroll
    for (int q = 0; q < 9; ++q) {
      Z[q] = ZA[q] + ZB[q];
      fr += Z[q] * Z[q];
    }
    float inv = 1.0f / (fr + 1.0f);
#pragma unroll
    for (int q = 0; q < 9; ++q) Z[q] *= inv;
    decomp3(Z, zc);
  } else {
#pragma unroll
    for (int q = 0; q < 9; ++q) zc[q] = 0.0f;
  }
  float* sz = &s_z[(nl * CCH + ch) * 9];
#pragma unroll
  for (int q = 0; q < 9; ++q) sz[q] = zc[q];
  __syncthreads();

  int o = ch;
  float acc[9] = {0, 0, 0, 0, 0, 0, 0, 0, 0};
  for (int c = 0; c < CCH; ++c) {
    float wi = s_WI[c * CCH + o];
    float wa = s_WA[c * CCH + o];
    float wsv = s_WS[c * CCH + o];
    const float* pc = &s_z[(nl * CCH + c) * 9];
    acc[0] += wi * pc[0];
    acc[1] += wa * pc[1]; acc[2] += wa * pc[2]; acc[3] += wa * pc[3];
    acc[4] += wsv * pc[4]; acc[5] += wsv * pc[5]; acc[6] += wsv * pc[6];
    acc[7] += wsv * pc[7]; acc[8] += wsv * pc[8];
  }
  if (n < N) {
    float P[9], PP[9];
    recon3(acc, P);
    mm3(P, P, PP);
    float* op = out + ((size_t)n * CCH + o) * 9;
#pragma unroll
    for (int q = 0; q < 9; ++q) op[q] = P[q] + PP[q];
  }
}

// ---------------------------------------------------------------------------
extern "C" void kernel_launch(void* const* d_in, const int* in_sizes, int n_in,
                              void* d_out, int out_size, void* d_ws,
                              size_t ws_size, hipStream_t stream) {
  const float* X = (const float*)d_in[0];
  const int* eidx = (const int*)d_in[1];
  const float* r = (const float*)d_in[2];
  const float* WIpre = (const float*)d_in[3];
  const float* WApre = (const float*)d_in[4];
  const float* WSpre = (const float*)d_in[5];
  const float* WIpost = (const float*)d_in[6];
  const float* WApost = (const float*)d_in[7];
  const float* WSpost = (const float*)d_in[8];
  const float* W1 = (const float*)d_in[9];
  const float* b1 = (const float*)d_in[10];
  const float* W2 = (const float*)d_in[11];
  const float* b2 = (const float*)d_in[12];

  int N = in_sizes[0] / (CCH * 9);
  int E = in_sizes[2];
  size_t NC9 = (size_t)N * CCH * 9;

  float* ws = (float*)d_ws;
  float* Part = ws;            // compact irreducible parts of X
  float* Y0c = ws + NC9;       // pre-mixed Y (compact)
  float* Mb = ws + 2 * NC9;    // scatter accumulator (compact)

  hipMemsetAsync(Mb, 0, NC9 * sizeof(float), stream);

  pre_kernel<<<(N + 3) / 4, 128, 0, stream>>>(X, WIpre, WApre, WSpre, Part,
                                              Y0c, N);
  edge_kernel<<<(E + 15) / 16, 128, 0, stream>>>(r, eidx, W1, b1, W2, b2, Part,
                                                 Mb, E);
  post_kernel<<<(N + 3) / 4, 128, 0, stream>>>(Y0c, Mb, WIpost, WApost, WSpost,
                                               (float*)d_out, N);
}